// Encoder_trace_627065225477
// MI455X (gfx1250) — compile-verified
//
#include <hip/hip_runtime.h>
#include <hip/hip_bf16.h>

typedef __attribute__((ext_vector_type(2))) float v2f;
typedef __attribute__((ext_vector_type(4))) float v4f;
typedef __attribute__((ext_vector_type(8))) float v8f;

#define LDS_W 36   // padded row stride for 32-wide K panels (144B: 16B aligned, conflict-free)

// ---------------------------------------------------------------------------
// Generic C = A(M,K) @ W(N,K)^T + bias, fp32 WMMA 16x16x4.
// Block = 256 threads (8 waves), 64x64 output tile, K staged 32 at a time.
// Wave w: msub = w&3 (16 rows), npair = w>>2 -> n-subtiles {2*npair, 2*npair+1}.
// ---------------------------------------------------------------------------
__global__ __launch_bounds__(256) void gemm_wmma_bias(
    const float* __restrict__ A, const float* __restrict__ W,
    const float* __restrict__ bias, float* __restrict__ C,
    int M, int N, int K)
{
  extern __shared__ float smem[];
  float* As = smem;                 // 64 x LDS_W
  float* Ws = smem + 64 * LDS_W;    // 64 x LDS_W

  const int tid  = threadIdx.x;
  const int wave = tid >> 5;
  const int lane = tid & 31;
  const int half = lane >> 4;
  const int l16  = lane & 15;
  const int msub  = wave & 3;
  const int npair = wave >> 2;

  const int row0 = blockIdx.x * 64;
  const int col0 = blockIdx.y * 64;

  v8f acc0 = {};
  v8f acc1 = {};

  for (int kk = 0; kk < K; kk += 32) {
    // stage 64x32 A panel and 64x32 W panel (512 float4 each, 2 per thread)
    #pragma unroll
    for (int rep = 0; rep < 2; ++rep) {
      int li = tid + rep * 256;      // 0..511
      int r  = li >> 3;
      int c4 = li & 7;
      *(v4f*)(As + r * LDS_W + c4 * 4) =
          *(const v4f*)(A + (size_t)(row0 + r) * K + kk + c4 * 4);
      *(v4f*)(Ws + r * LDS_W + c4 * 4) =
          *(const v4f*)(W + (size_t)(col0 + r) * K + kk + c4 * 4);
    }
    if (kk + 32 < K) {
      // locality 3 -> WGP-scope prefetch: pull next K panel into all cache levels
      __builtin_prefetch(A + (size_t)(row0 + (tid >> 2)) * K + kk + 32, 0, 3);
      __builtin_prefetch(W + (size_t)(col0 + (tid >> 2)) * K + kk + 32, 0, 3);
    }
    __syncthreads();

    #pragma unroll
    for (int ks = 0; ks < 32; ks += 4) {
      const int kidx = ks + 2 * half;
      v2f a  = *(const v2f*)(As + (msub * 16 + l16) * LDS_W + kidx);
      v2f b0 = *(const v2f*)(Ws + ((2 * npair) * 16 + l16) * LDS_W + kidx);
      v2f b1 = *(const v2f*)(Ws + ((2 * npair + 1) * 16 + l16) * LDS_W + kidx);
      acc0 = __builtin_amdgcn_wmma_f32_16x16x4_f32(false, a, false, b0,
                                                   (short)0, acc0, false, false);
      acc1 = __builtin_amdgcn_wmma_f32_16x16x4_f32(false, a, false, b1,
                                                   (short)0, acc1, false, false);
    }
    __syncthreads();
  }

  const int rbase = row0 + msub * 16 + 8 * half;
  const int cb0 = col0 + (2 * npair) * 16 + l16;
  const int cb1 = cb0 + 16;
  const float bz0 = bias ? bias[cb0] : 0.0f;
  const float bz1 = bias ? bias[cb1] : 0.0f;
  #pragma unroll
  for (int i = 0; i < 8; ++i) {
    C[(size_t)(rbase + i) * N + cb0] = acc0[i] + bz0;
    C[(size_t)(rbase + i) * N + cb1] = acc1[i] + bz1;
  }
}

// ---------------------------------------------------------------------------
// GCN edge fixup: x_time = h2/deg + coef*h2[src] (dst in 1..4) + b_gcn
// EDGE_SRC=[0,1,2,3], EDGE_DST=[1,2,3,4], deg[f]=2 for f in 1..4 else 1.
// ---------------------------------------------------------------------------
__global__ __launch_bounds__(256) void gcn_fix(
    const float* __restrict__ h2, const float* __restrict__ bg,
    float* __restrict__ xt, int F, int H)
{
  size_t idx = (size_t)blockIdx.x * 256 + threadIdx.x;
  int h = (int)(idx % H);
  size_t bf = idx / H;
  int f = (int)(bf % F);
  float degd = (f >= 1 && f <= 4) ? 2.0f : 1.0f;
  float v = h2[idx] * (1.0f / degd);
  if (f >= 1 && f <= 4) {
    int fs = f - 1;
    float degs = (fs >= 1 && fs <= 4) ? 2.0f : 1.0f;
    v += rsqrtf(degs * degd) * h2[idx - H];
  }
  xt[idx] = v + bg[h];
}

// ---------------------------------------------------------------------------
// weT[v*H + h] = wemb[h*V + v]   (word_embedding (H,V) -> (V,H))
// ---------------------------------------------------------------------------
__global__ __launch_bounds__(256) void transpose_we(
    const float* __restrict__ wemb, float* __restrict__ weT, int V, int H)
{
  int idx = blockIdx.x * 256 + threadIdx.x;   // idx = v*H + h
  int h = idx % H;
  int v = idx / H;
  weT[idx] = wemb[h * V + v];
}

// ---------------------------------------------------------------------------
// Fused attention per (batch, head, 32-query chunk). NKEY=256, dh=64.
// Phase 1: S(32x256) in regs via WMMA over 4 key tiles of 64.
// Softmax: shfl within 16-lane groups + LDS reduction across 4 column quads.
// Phase 2: O = P @ V via WMMA, V staged transposed in LDS.
// Wave w: msub = w&1 (16 q-rows), quad = w>>1 (16-wide column quadrant).
// LDS: P 32x260 (overlays q 32x68, k 64x68), vT 64x68, red 32x4  = 12800 f.
// ---------------------------------------------------------------------------
__global__ __launch_bounds__(256) void attn_wmma(
    const float* __restrict__ Q, const float* __restrict__ Km,
    const float* __restrict__ Vm, float* __restrict__ O,
    int F, int H, float scale)
{
  extern __shared__ float smem[];
  float* Pl  = smem;                 // 32 x 260 (phase 2)
  float* ql  = smem;                 // 32 x 68  (phase 1)
  float* kl  = smem + 32 * 68;       // 64 x 68  (phase 1)
  float* vl  = smem + 32 * 260;      // 64 x 68  (vT: d x key)
  float* red = smem + 32 * 260 + 64 * 68;  // 32 x 4

  const int tid  = threadIdx.x;
  const int wave = tid >> 5;
  const int lane = tid & 31;
  const int half = lane >> 4;
  const int l16  = lane & 15;
  const int msub = wave & 1;
  const int quad = wave >> 1;

  const int qc = blockIdx.x, head = blockIdx.y, b = blockIdx.z;
  const int dh = 64;
  const size_t qrow0 = (size_t)b * F + qc * 32;
  const int hoff = head * dh;

  // stage q chunk 32x64
  for (int li = tid; li < 32 * 16; li += 256) {
    int r = li >> 4, c4 = li & 15;
    *(v4f*)(ql + r * 68 + c4 * 4) =
        *(const v4f*)(Q + (qrow0 + r) * H + hoff + c4 * 4);
  }

  v8f acc[4] = {};
  #pragma unroll
  for (int kt = 0; kt < 4; ++kt) {
    __syncthreads();
    for (int li = tid; li < 64 * 16; li += 256) {
      int r = li >> 4, c4 = li & 15;
      *(v4f*)(kl + r * 68 + c4 * 4) =
          *(const v4f*)(Km + (size_t)(kt * 64 + r) * H + hoff + c4 * 4);
    }
    __syncthreads();
    #pragma unroll
    for (int ks = 0; ks < 64; ks += 4) {
      const int kidx = ks + 2 * half;
      v2f a  = *(const v2f*)(ql + (msub * 16 + l16) * 68 + kidx);
      v2f bf = *(const v2f*)(kl + (quad * 16 + l16) * 68 + kidx);
      acc[kt] = __builtin_amdgcn_wmma_f32_16x16x4_f32(false, a, false, bf,
                                                      (short)0, acc[kt], false, false);
    }
  }
  __syncthreads();

  // ---- softmax over 256 keys per row ----
  #pragma unroll
  for (int j = 0; j < 4; ++j)
    #pragma unroll
    for (int i = 0; i < 8; ++i) acc[j][i] *= scale;

  float pm[8];
  #pragma unroll
  for (int i = 0; i < 8; ++i) pm[i] = -3.4e38f;
  #pragma unroll
  for (int j = 0; j < 4; ++j)
    #pragma unroll
    for (int i = 0; i < 8; ++i) pm[i] = fmaxf(pm[i], acc[j][i]);
  #pragma unroll
  for (int m = 8; m >= 1; m >>= 1)
    #pragma unroll
    for (int i = 0; i < 8; ++i) pm[i] = fmaxf(pm[i], __shfl_xor(pm[i], m, 32));

  const int rbase = msub * 16 + 8 * half;   // local q-row base (0..31)
  if (l16 == 0) {
    #pragma unroll
    for (int i = 0; i < 8; ++i) red[(rbase + i) * 4 + quad] = pm[i];
  }
  __syncthreads();
  float rowm[8];
  #pragma unroll
  for (int i = 0; i < 8; ++i) {
    const float* rr = red + (rbase + i) * 4;
    rowm[i] = fmaxf(fmaxf(rr[0], rr[1]), fmaxf(rr[2], rr[3]));
  }
  __syncthreads();

  float ps[8];
  #pragma unroll
  for (int i = 0; i < 8; ++i) ps[i] = 0.0f;
  #pragma unroll
  for (int j = 0; j < 4; ++j)
    #pragma unroll
    for (int i = 0; i < 8; ++i) {
      float e = __expf(acc[j][i] - rowm[i]);
      acc[j][i] = e;
      ps[i] += e;
    }
  #pragma unroll
  for (int m = 8; m >= 1; m >>= 1)
    #pragma unroll
    for (int i = 0; i < 8; ++i) ps[i] += __shfl_xor(ps[i], m, 32);
  if (l16 == 0) {
    #pragma unroll
    for (int i = 0; i < 8; ++i) red[(rbase + i) * 4 + quad] = ps[i];
  }
  __syncthreads();
  float rinv[8];
  #pragma unroll
  for (int i = 0; i < 8; ++i) {
    const float* rr = red + (rbase + i) * 4;
    rinv[i] = 1.0f / (rr[0] + rr[1] + rr[2] + rr[3]);
  }

  // store normalized P (32 x 256) to LDS (overwrites q/k panels — all reads done)
  #pragma unroll
  for (int j = 0; j < 4; ++j)
    #pragma unroll
    for (int i = 0; i < 8; ++i)
      Pl[(rbase + i) * 260 + j * 64 + quad * 16 + l16] = acc[j][i] * rinv[i];
  __syncthreads();

  // ---- O = P @ V ----
  v8f oacc = {};
  #pragma unroll
  for (int kt = 0; kt < 4; ++kt) {
    for (int li = tid; li < 4096; li += 256) {
      int keyrel = li >> 6, d = li & 63;
      vl[d * 68 + keyrel] = Vm[(size_t)(kt * 64 + keyrel) * H + hoff + d];
    }
    __syncthreads();
    #pragma unroll
    for (int ks = 0; ks < 64; ks += 4) {
      const int kidx = ks + 2 * half;
      v2f a  = *(const v2f*)(Pl + (msub * 16 + l16) * 260 + kt * 64 + kidx);
      v2f bf = *(const v2f*)(vl + (quad * 16 + l16) * 68 + kidx);
      oacc = __builtin_amdgcn_wmma_f32_16x16x4_f32(false, a, false, bf,
                                                   (short)0, oacc, false, false);
    }
    __syncthreads();
  }

  #pragma unroll
  for (int i = 0; i < 8; ++i) {
    int r = rbase + i;
    int d = quad * 16 + l16;
    O[(qrow0 + r) * H + hoff + d] = oacc[i];
  }
}

// ---------------------------------------------------------------------------
extern "C" void kernel_launch(void* const* d_in, const int* in_sizes, int n_in,
                              void* d_out, int out_size, void* d_ws, size_t ws_size,
                              hipStream_t stream) {
  (void)in_sizes; (void)n_in; (void)out_size; (void)ws_size;

  constexpr int B = 64, F = 512, D_IN = 256, H = 768, NH = 12, V = 256;
  constexpr int M = B * F;                 // 32768
  const float scale = 0.125f;              // (H/NH=64)^-0.5

  const float* x     = (const float*)d_in[0];
  const float* wemb  = (const float*)d_in[1];
  const float* W_lin = (const float*)d_in[2];
  const float* b_lin = (const float*)d_in[3];
  const float* W_gcn = (const float*)d_in[4];
  const float* b_gcn = (const float*)d_in[5];
  const float* in_w  = (const float*)d_in[6];
  const float* in_b  = (const float*)d_in[7];
  const float* out_w = (const float*)d_in[8];
  const float* out_b = (const float*)d_in[9];

  float* out = (float*)d_out;
  float* ws  = (float*)d_ws;

  float* buf0  = ws;                               // h1, then q
  float* buf1  = buf0 + (size_t)M * H;             // h2, then o
  float* kbuf  = buf1 + (size_t)M * H;
  float* vbuf  = kbuf + (size_t)V * H;
  float* webuf = vbuf + (size_t)V * H;

  float* x_time = out;                             // output #1 (also Q input)
  float* x_out  = out + (size_t)M * H;             // output #2

  const size_t gemm_lds = 2 * 64 * LDS_W * sizeof(float);          // 18432 B
  const size_t attn_lds = (32 * 260 + 64 * 68 + 32 * 4) * sizeof(float); // 51200 B

  // 1. h1 = x @ W_lin^T + b_lin
  gemm_wmma_bias<<<dim3(M / 64, H / 64), 256, gemm_lds, stream>>>(
      x, W_lin, b_lin, buf0, M, H, D_IN);
  // 2. h2 = h1 @ W_gcn^T
  gemm_wmma_bias<<<dim3(M / 64, H / 64), 256, gemm_lds, stream>>>(
      buf0, W_gcn, nullptr, buf1, M, H, H);
  // 3. x_time = gcn(h2) + b_gcn   (output #1)
  gcn_fix<<<(size_t)M * H / 256, 256, 0, stream>>>(buf1, b_gcn, x_time, F, H);
  // 4. we = word_embedding^T
  transpose_we<<<(V * H) / 256, 256, 0, stream>>>(wemb, webuf, V, H);
  // 5. k = we @ Wk^T + bk ; v = we @ Wv^T + bv   (batch-invariant: computed once)
  gemm_wmma_bias<<<dim3(V / 64, H / 64), 256, gemm_lds, stream>>>(
      webuf, in_w + (size_t)H * H, in_b + H, kbuf, V, H, H);
  gemm_wmma_bias<<<dim3(V / 64, H / 64), 256, gemm_lds, stream>>>(
      webuf, in_w + (size_t)2 * H * H, in_b + 2 * H, vbuf, V, H, H);
  // 6. q = x_time @ Wq^T + bq
  gemm_wmma_bias<<<dim3(M / 64, H / 64), 256, gemm_lds, stream>>>(
      x_time, in_w, in_b, buf0, M, H, H);
  // 7. o = softmax(q k^T * scale) v
  attn_wmma<<<dim3(F / 32, NH, B), 256, attn_lds, stream>>>(
      buf0, kbuf, vbuf, buf1, F, H, scale);
  // 8. x_out = o @ out_proj_w^T + out_proj_b   (output #2)
  gemm_wmma_bias<<<dim3(M / 64, H / 64), 256, gemm_lds, stream>>>(
      buf1, out_w, out_b, x_out, M, H, H);
}